// GNN_layer_8873402433731
// MI455X (gfx1250) — compile-verified
//
#include <hip/hip_runtime.h>

typedef __attribute__((ext_vector_type(2))) float v2f;
typedef __attribute__((ext_vector_type(8))) float v8f;

// Stable mish: x * tanh(softplus(x)), softplus(x) = max(x,0) + log1p(exp(-|x|))
__device__ __forceinline__ float mish_f(float x) {
    float sp = fmaxf(x, 0.0f) + log1pf(__expf(-fabsf(x)));
    return x * tanhf(sp);
}

// ---------------------------------------------------------------------------
// Mean over G=16 groups: BB_feature [4096,16,128] -> bb_mean [4096,128]
// ---------------------------------------------------------------------------
__global__ __launch_bounds__(256) void bb_mean_kernel(const float* __restrict__ bb,
                                                      float* __restrict__ mean) {
    int idx = blockIdx.x * blockDim.x + threadIdx.x;   // 0 .. 4096*128-1
    int b = idx >> 7;          // batch
    int d = idx & 127;         // feature
    const float* p = bb + ((size_t)b << 11) + d;       // b * (16*128)
    float s = 0.0f;
#pragma unroll
    for (int g = 0; g < 16; ++g) s += p[g << 7];
    mean[idx] = s * (1.0f / 16.0f);
}

// ---------------------------------------------------------------------------
// WMMA fp32 GEMM: out[M,N] = mish(A[M,K] * W[N,K]^T + bias[N]) (+ addv[m/16, N])
// One wave computes a (16*MR) x 64 block: MR M-tiles x 4 N-tiles.
// B fragment reused across MR WMMAs; A fragments reused across 4 N-tiles.
// grid = (M/(16*MR)) * (N/64) / 8 blocks of 256 threads (8 waves).
// ---------------------------------------------------------------------------
template <int MR, bool HAS_ADD>
__global__ __launch_bounds__(256) void wmma_gemm_mish(
    const float* __restrict__ A, const float* __restrict__ W,
    const float* __restrict__ bias, const float* __restrict__ addv,
    float* __restrict__ out, int M, int N, int K)
{
    const int lane = threadIdx.x & 31;
    const int half = lane >> 4;
    const int lrow = lane & 15;
    const int Mg   = M / (16 * MR);

    int w  = blockIdx.x * (blockDim.x >> 5) + (threadIdx.x >> 5);
    int mg = w % Mg;
    int nq = w / Mg;
    const int m0 = mg * (16 * MR);
    const int n0 = nq << 6;

    const float* Ap = A + (size_t)(m0 + lrow) * K + 2 * half;
    const float* Wp = W + (size_t)(n0 + lrow) * K + 2 * half;

    v8f acc[MR][4] = {};
#pragma unroll 2
    for (int k0 = 0; k0 < K; k0 += 4) {
        v2f a[MR];
#pragma unroll
        for (int i = 0; i < MR; ++i)
            a[i] = *(const v2f*)(Ap + (size_t)(i * 16) * K + k0);
#pragma unroll
        for (int j = 0; j < 4; ++j) {
            v2f b = *(const v2f*)(Wp + (size_t)(j * 16) * K + k0);
#pragma unroll
            for (int i = 0; i < MR; ++i)
                acc[i][j] = __builtin_amdgcn_wmma_f32_16x16x4_f32(
                    false, a[i], false, b, (short)0, acc[i][j], false, false);
        }
    }

#pragma unroll
    for (int j = 0; j < 4; ++j) {
        int col  = n0 + j * 16 + lrow;
        float bs = bias[col];
#pragma unroll
        for (int i = 0; i < MR; ++i) {
#pragma unroll
            for (int r = 0; r < 8; ++r) {
                int row = m0 + i * 16 + r + 8 * half;
                float v = mish_f(acc[i][j][r] + bs);
                if (HAS_ADD) v += addv[(size_t)(row >> 4) * N + col];
                out[(size_t)row * N + col] = v;
            }
        }
    }
}

// ---------------------------------------------------------------------------
// Fused dual GEMM: out = mish(A1*W1^T + b1) + mish(A2*W2^T + b2), both [M,N]
// Same (16*MR) x 64 per-wave blocking, two accumulator sets.
// ---------------------------------------------------------------------------
template <int MR>
__global__ __launch_bounds__(256) void wmma_gemm_dual_mish(
    const float* __restrict__ A1, const float* __restrict__ W1,
    const float* __restrict__ bias1, int K1,
    const float* __restrict__ A2, const float* __restrict__ W2,
    const float* __restrict__ bias2, int K2,
    float* __restrict__ out, int M, int N)
{
    const int lane = threadIdx.x & 31;
    const int half = lane >> 4;
    const int lrow = lane & 15;
    const int Mg   = M / (16 * MR);

    int w  = blockIdx.x * (blockDim.x >> 5) + (threadIdx.x >> 5);
    int mg = w % Mg;
    int nq = w / Mg;
    const int m0 = mg * (16 * MR);
    const int n0 = nq << 6;

    v8f acc1[MR][4] = {};
    v8f acc2[MR][4] = {};

    {
        const float* Ap = A1 + (size_t)(m0 + lrow) * K1 + 2 * half;
        const float* Wp = W1 + (size_t)(n0 + lrow) * K1 + 2 * half;
#pragma unroll 2
        for (int k0 = 0; k0 < K1; k0 += 4) {
            v2f a[MR];
#pragma unroll
            for (int i = 0; i < MR; ++i)
                a[i] = *(const v2f*)(Ap + (size_t)(i * 16) * K1 + k0);
#pragma unroll
            for (int j = 0; j < 4; ++j) {
                v2f b = *(const v2f*)(Wp + (size_t)(j * 16) * K1 + k0);
#pragma unroll
                for (int i = 0; i < MR; ++i)
                    acc1[i][j] = __builtin_amdgcn_wmma_f32_16x16x4_f32(
                        false, a[i], false, b, (short)0, acc1[i][j], false, false);
            }
        }
    }
    {
        const float* Ap = A2 + (size_t)(m0 + lrow) * K2 + 2 * half;
        const float* Wp = W2 + (size_t)(n0 + lrow) * K2 + 2 * half;
#pragma unroll 2
        for (int k0 = 0; k0 < K2; k0 += 4) {
            v2f a[MR];
#pragma unroll
            for (int i = 0; i < MR; ++i)
                a[i] = *(const v2f*)(Ap + (size_t)(i * 16) * K2 + k0);
#pragma unroll
            for (int j = 0; j < 4; ++j) {
                v2f b = *(const v2f*)(Wp + (size_t)(j * 16) * K2 + k0);
#pragma unroll
                for (int i = 0; i < MR; ++i)
                    acc2[i][j] = __builtin_amdgcn_wmma_f32_16x16x4_f32(
                        false, a[i], false, b, (short)0, acc2[i][j], false, false);
            }
        }
    }

#pragma unroll
    for (int j = 0; j < 4; ++j) {
        int col   = n0 + j * 16 + lrow;
        float b1v = bias1[col];
        float b2v = bias2[col];
#pragma unroll
        for (int i = 0; i < MR; ++i) {
#pragma unroll
            for (int r = 0; r < 8; ++r) {
                int row = m0 + i * 16 + r + 8 * half;
                float v = mish_f(acc1[i][j][r] + b1v) + mish_f(acc2[i][j][r] + b2v);
                out[(size_t)row * N + col] = v;
            }
        }
    }
}

// ---------------------------------------------------------------------------
extern "C" void kernel_launch(void* const* d_in, const int* in_sizes, int n_in,
                              void* d_out, int out_size, void* d_ws, size_t ws_size,
                              hipStream_t stream) {
    // Model constants
    const int B      = 4096;
    const int G      = 16;
    const int BB_DIM = 128;
    const int RF_DIM = 1024;
    const int MR     = 2;      // M-tiles per wave

    const float* BB_feature = (const float*)d_in[0];  // [B,G,BB_DIM]
    const float* RF_feature = (const float*)d_in[1];  // [B,RF_DIM]
    const float* W_rfbb     = (const float*)d_in[2];  // [BB_DIM,RF_DIM]
    const float* b_rfbb     = (const float*)d_in[3];
    const float* W_rfrf     = (const float*)d_in[4];  // [RF_DIM,RF_DIM]
    const float* b_rfrf     = (const float*)d_in[5];
    const float* W_bbrf     = (const float*)d_in[6];  // [RF_DIM,BB_DIM]
    const float* b_bbrf     = (const float*)d_in[7];
    const float* W_bbbb     = (const float*)d_in[8];  // [BB_DIM,BB_DIM]
    const float* b_bbbb     = (const float*)d_in[9];

    float* out_BB = (float*)d_out;                          // [B,G,BB_DIM]
    float* out_RF = out_BB + (size_t)B * G * BB_DIM;        // [B,RF_DIM]

    float* bb_mean  = (float*)d_ws;                         // [B,BB_DIM]
    float* rf_to_bb = bb_mean + (size_t)B * BB_DIM;         // [B,BB_DIM]

    // 1) BB mean over groups
    bb_mean_kernel<<<(B * BB_DIM) / 256, 256, 0, stream>>>(BB_feature, bb_mean);

    // 2) RF_to_BB = mish(RF * W_rfbb^T + b_rfbb)   [B,128], K=1024
    {
        int M = B, N = BB_DIM, K = RF_DIM;
        int blocks = (M / (16 * MR)) * (N / 64) / 8;        // 32
        wmma_gemm_mish<MR, false><<<blocks, 256, 0, stream>>>(
            RF_feature, W_rfbb, b_rfbb, nullptr, rf_to_bb, M, N, K);
    }

    // 3) new_RF = mish(RF*W_rfrf^T+b) + mish(bb_mean*W_bbrf^T+b)  [B,1024]
    {
        int M = B, N = RF_DIM;
        int blocks = (M / (16 * MR)) * (N / 64) / 8;        // 256
        wmma_gemm_dual_mish<MR><<<blocks, 256, 0, stream>>>(
            RF_feature, W_rfrf, b_rfrf, RF_DIM,
            bb_mean,    W_bbrf, b_bbrf, BB_DIM,
            out_RF, M, N);
    }

    // 4) new_BB = mish(BB_feature*W_bbbb^T + b) + RF_to_BB[b]  [B*G,128], K=128
    {
        int M = B * G, N = BB_DIM, K = BB_DIM;
        int blocks = (M / (16 * MR)) * (N / 64) / 8;        // 512
        wmma_gemm_mish<MR, true><<<blocks, 256, 0, stream>>>(
            BB_feature, W_bbbb, b_bbbb, rf_to_bb, out_BB, M, N, K);
    }
}